// GraphAttentionLayer_15187004358827
// MI455X (gfx1250) — compile-verified
//
#include <hip/hip_runtime.h>
#include <math.h>

typedef float v2f __attribute__((ext_vector_type(2)));
typedef float v8f __attribute__((ext_vector_type(8)));

#define Nn   3072
#define Pn   4
#define INF_ 256
#define Ff   64
#define OUTW 256          // P*F, width of concatenated output
#define ALPHA 0.2f

// ---------------- workspace layout (floats) ----------------
#define WS_WH   0                       // 3072*64   = 196608
#define WS_WH1  196608                  // 3072
#define WS_WH2  199680                  // 3072
#define WS_RS   202752                  // 4*3072
#define WS_CS   215040                  // 4*3072
#define WS_U    227328                  // 4*3072
#define WS_V    239616                  // 4*3072
#define WS_INVS 251904                  // 4

// ================= K1: Wh = h@W, Wh1 = Wh@a[:F], Wh2 = Wh@a[F:] =================
__global__ __launch_bounds__(256)
void k1_wh(const float* __restrict__ h, const float* __restrict__ W,
           const float* __restrict__ a, float* __restrict__ Wh,
           float* __restrict__ wh1, float* __restrict__ wh2) {
    __shared__ float hs[4 * INF_];
    __shared__ float s1[256], s2[256];
    const int t  = threadIdx.x;
    const int i0 = blockIdx.x * 4;
    // stage 4 rows of h (4*256 floats) into LDS
    {
        int r = t >> 6;            // 0..3
        int c = (t & 63) * 4;      // 0..252
        float4 hv = *(const float4*)&h[(size_t)(i0 + r) * INF_ + c];
        hs[r * INF_ + c + 0] = hv.x; hs[r * INF_ + c + 1] = hv.y;
        hs[r * INF_ + c + 2] = hv.z; hs[r * INF_ + c + 3] = hv.w;
    }
    __syncthreads();
    const int r = t >> 6;          // row within block
    const int f = t & 63;          // output feature
    float acc = 0.f;
#pragma unroll 8
    for (int k = 0; k < INF_; ++k)
        acc = fmaf(hs[r * INF_ + k], W[k * Ff + f], acc);
    Wh[(size_t)(i0 + r) * Ff + f] = acc;
    s1[t] = acc * a[f];
    s2[t] = acc * a[Ff + f];
    __syncthreads();
    for (int off = 32; off > 0; off >>= 1) {
        if (f < off) { s1[t] += s1[t + off]; s2[t] += s2[t + off]; }
        __syncthreads();
    }
    if (f == 0) { wh1[i0 + r] = s1[t]; wh2[i0 + r] = s2[t]; }
}

// ========= K2: e = leaky_relu(wh1_i + wh2_j) * edge, plus row sums =========
__global__ __launch_bounds__(256)
void k2_e_rowsum(const float* __restrict__ edge, const float* __restrict__ wh1,
                 const float* __restrict__ wh2, float* __restrict__ out_e,
                 float* __restrict__ rs) {
    __shared__ float red[256];
    const int i = blockIdx.x, p = blockIdx.y, t = threadIdx.x;
    const float w1 = wh1[i];
    const float* erow = edge  + ((size_t)p * Nn + i) * (size_t)Nn;
    float*       orow = out_e + ((size_t)p * Nn + i) * (size_t)Nn;
    float sum = 0.f;
    for (int jq = t; jq < Nn / 4; jq += 256) {
        float4 ea = *(const float4*)&erow[jq * 4];
        float4 w2 = *(const float4*)&wh2[jq * 4];
        float4 ev;
        float s;
        s = w1 + w2.x; ev.x = (s > 0.f ? s : ALPHA * s) * ea.x;
        s = w1 + w2.y; ev.y = (s > 0.f ? s : ALPHA * s) * ea.y;
        s = w1 + w2.z; ev.z = (s > 0.f ? s : ALPHA * s) * ea.z;
        s = w1 + w2.w; ev.w = (s > 0.f ? s : ALPHA * s) * ea.w;
        *(float4*)&orow[jq * 4] = ev;
        sum += ev.x + ev.y + ev.z + ev.w;
    }
    red[t] = sum;
    __syncthreads();
    for (int off = 128; off > 0; off >>= 1) {
        if (t < off) red[t] += red[t + off];
        __syncthreads();
    }
    if (t == 0) rs[p * Nn + i] = red[0];
}

// ================= K3: column sums of e (deterministic, coalesced) =================
__global__ __launch_bounds__(256)
void k3_colsum(const float* __restrict__ out_e, float* __restrict__ cs) {
    const int j = blockIdx.x * 256 + threadIdx.x;
    const int p = blockIdx.y;
    const float* ep = out_e + (size_t)p * Nn * Nn;
    float s = 0.f;
#pragma unroll 8
    for (int i = 0; i < Nn; ++i)
        s += ep[(size_t)i * Nn + j];
    cs[p * Nn + j] = s;
}

// ====== K4: per-slice scalars; u_i = (lam-rs_i)/r, v_j = lam-cs_j, invS = 1/lam ======
// (closed form: S = sum_i tt[i][0] = cs0 + (lam-cs0)*(N*lam - T)/r = lam)
__global__ __launch_bounds__(256)
void k4_scalars(const float* __restrict__ rs, const float* __restrict__ cs,
                float* __restrict__ u, float* __restrict__ v,
                float* __restrict__ invS) {
    __shared__ float smr[256], ssm[256], smc[256];
    __shared__ float sc[2];
    const int p = blockIdx.x, t = threadIdx.x;
    float mr = -3.4e38f, mc = -3.4e38f, sm = 0.f;
    for (int i = t; i < Nn; i += 256) {
        float rv = rs[p * Nn + i];
        mr = fmaxf(mr, rv); sm += rv;
        mc = fmaxf(mc, cs[p * Nn + i]);
    }
    smr[t] = mr; ssm[t] = sm; smc[t] = mc;
    __syncthreads();
    for (int off = 128; off > 0; off >>= 1) {
        if (t < off) {
            smr[t] = fmaxf(smr[t], smr[t + off]);
            smc[t] = fmaxf(smc[t], smc[t + off]);
            ssm[t] += ssm[t + off];
        }
        __syncthreads();
    }
    if (t == 0) {
        float lam = fmaxf(smr[0], smc[0]);
        float r   = (float)Nn * lam - ssm[0];
        sc[0] = lam; sc[1] = 1.f / r;
        invS[p] = 1.f / lam;
    }
    __syncthreads();
    const float lam = sc[0], invr = sc[1];
    for (int i = t; i < Nn; i += 256) {
        u[p * Nn + i] = (lam - rs[p * Nn + i]) * invr;
        v[p * Nn + i] = lam - cs[p * Nn + i];
    }
}

// ============ K5: out1 = elu(att @ Wh), att built on the fly from e ============
// Block = 256 threads (8 waves), computes 64 rows x 64 cols for one slice p.
// Wave w: row-tile = w>>1 (16 rows), col-tiles {2*(w&1), 2*(w&1)+1}.
// K stepped by 4 via V_WMMA_F32_16X16X4_F32; A/B staged through padded LDS.
#define KC  64
#define LDA 68
__global__ __launch_bounds__(256)
void k5_gemm_wmma(const float* __restrict__ out_e, const float* __restrict__ Wh,
                  const float* __restrict__ u, const float* __restrict__ v,
                  const float* __restrict__ invS, float* __restrict__ out1) {
    __shared__ float sA[64 * LDA];
    __shared__ float sB[64 * LDA];
    const int t  = threadIdx.x;
    const int i0 = blockIdx.x * 64;
    const int p  = blockIdx.y;
    const float* ep  = out_e + (size_t)p * Nn * Nn;
    const float invs = invS[p];

    // loader mapping: each thread owns row lr (0..63), 16-col group cg
    const int lr = t >> 2;
    const int cg = (t & 3) * 16;
    const float urow = u[p * Nn + i0 + lr];

    // compute mapping
    const int w     = t >> 5;
    const int lane  = t & 31;
    const int rtile = w >> 1;
    const int c0    = (w & 1) * 2;
    const int arow  = rtile * 16 + (lane & 15);
    const int bn0   = c0 * 16 + (lane & 15);
    const int bn1   = bn0 + 16;
    const int khalf = (lane >> 4) * 2;   // lanes 0-15: K=0/1, lanes 16-31: K=2/3

    v8f acc0 = {0.f, 0.f, 0.f, 0.f, 0.f, 0.f, 0.f, 0.f};
    v8f acc1 = {0.f, 0.f, 0.f, 0.f, 0.f, 0.f, 0.f, 0.f};

    for (int kc = 0; kc < Nn; kc += KC) {
        __syncthreads();   // protect LDS reuse from previous iteration's compute
        // stage A = attention tile (64 x KC), transform applied on the fly
        {
            const float* grow = &ep[(size_t)(i0 + lr) * Nn + kc + cg];
            const float* vrow = &v[p * Nn + kc + cg];
#pragma unroll
            for (int q = 0; q < 4; ++q) {
                float4 ev = *(const float4*)&grow[q * 4];
                float4 vv = *(const float4*)&vrow[q * 4];
                int base = lr * LDA + cg + q * 4;
                sA[base + 0] = ev.x > 0.f ? (ev.x + urow * vv.x) * invs : ev.x;
                sA[base + 1] = ev.y > 0.f ? (ev.y + urow * vv.y) * invs : ev.y;
                sA[base + 2] = ev.z > 0.f ? (ev.z + urow * vv.z) * invs : ev.z;
                sA[base + 3] = ev.w > 0.f ? (ev.w + urow * vv.w) * invs : ev.w;
            }
        }
        // stage B = Wh tile (KC x 64)
        {
            const float* brow = &Wh[(size_t)(kc + lr) * Ff + cg];
#pragma unroll
            for (int q = 0; q < 4; ++q) {
                float4 bv = *(const float4*)&brow[q * 4];
                int base = lr * LDA + cg + q * 4;
                sB[base + 0] = bv.x; sB[base + 1] = bv.y;
                sB[base + 2] = bv.z; sB[base + 3] = bv.w;
            }
        }
        __syncthreads();
#pragma unroll
        for (int k4 = 0; k4 < KC / 4; ++k4) {
            const int ak = k4 * 4 + khalf;
            v2f a = *(const v2f*)&sA[arow * LDA + ak];
            v2f b0, b1;
            b0.x = sB[ak * LDA + bn0];       b0.y = sB[(ak + 1) * LDA + bn0];
            b1.x = sB[ak * LDA + bn1];       b1.y = sB[(ak + 1) * LDA + bn1];
            acc0 = __builtin_amdgcn_wmma_f32_16x16x4_f32(
                false, a, false, b0, (short)0, acc0, false, false);
            acc1 = __builtin_amdgcn_wmma_f32_16x16x4_f32(
                false, a, false, b1, (short)0, acc1, false, false);
        }
    }
    // store with ELU; C/D layout: VGPR g -> M = g + 8*(lane>=16), N = lane&15
    const int n     = lane & 15;
    const int mbase = (lane >> 4) * 8;
#pragma unroll
    for (int g = 0; g < 8; ++g) {
        const int row = i0 + rtile * 16 + mbase + g;
        float x0 = acc0[g], x1 = acc1[g];
        out1[(size_t)row * OUTW + p * Ff + c0 * 16 + n]       = x0 > 0.f ? x0 : expm1f(x0);
        out1[(size_t)row * OUTW + p * Ff + (c0 + 1) * 16 + n] = x1 > 0.f ? x1 : expm1f(x1);
    }
}

// ================================ launcher ================================
extern "C" void kernel_launch(void* const* d_in, const int* in_sizes, int n_in,
                              void* d_out, int out_size, void* d_ws, size_t ws_size,
                              hipStream_t stream) {
    const float* h    = (const float*)d_in[0];   // [3072, 256]
    const float* edge = (const float*)d_in[1];   // [4, 3072, 3072]
    const float* W    = (const float*)d_in[2];   // [256, 64]
    const float* a    = (const float*)d_in[3];   // [128, 1]

    float* ws   = (float*)d_ws;
    float* Wh   = ws + WS_WH;
    float* wh1  = ws + WS_WH1;
    float* wh2  = ws + WS_WH2;
    float* rs   = ws + WS_RS;
    float* cs   = ws + WS_CS;
    float* u    = ws + WS_U;
    float* v    = ws + WS_V;
    float* invS = ws + WS_INVS;

    float* out1  = (float*)d_out;                 // elu(h_cat): 3072*256
    float* out_e = out1 + (size_t)Nn * OUTW;      // e: 4*3072*3072

    k1_wh        <<<dim3(Nn / 4),        256, 0, stream>>>(h, W, a, Wh, wh1, wh2);
    k2_e_rowsum  <<<dim3(Nn, Pn),        256, 0, stream>>>(edge, wh1, wh2, out_e, rs);
    k3_colsum    <<<dim3(Nn / 256, Pn),  256, 0, stream>>>(out_e, cs);
    k4_scalars   <<<dim3(Pn),            256, 0, stream>>>(rs, cs, u, v, invS);
    k5_gemm_wmma <<<dim3(Nn / 64, Pn),   256, 0, stream>>>(out_e, Wh, u, v, invS, out1);
}